// VectorQuantizer_44753559224674
// MI455X (gfx1250) — compile-verified
//
#include <hip/hip_runtime.h>

#define D_MODEL 512
#define N_BINS  8192
#define N_VEC   (8 * 4096)          // 32768 z-vectors
#define BETA    0.25f

#define LDS_STRIDE 516              // 512 + 4 floats pad -> conflict-free b64 LDS reads
#define M_TILE 64                   // z rows per workgroup
#define WG_THREADS 128              // 4 waves, each owns 16 rows
#define N_TILES (N_BINS / 16)       // 512 codebook tiles of 16 columns

typedef __attribute__((ext_vector_type(2))) float v2f;
typedef __attribute__((ext_vector_type(4))) float v4f;
typedef __attribute__((ext_vector_type(8))) float v8f;

// ---- CDNA5 async global->LDS helpers (ASYNCcnt path) ----------------------
__device__ __forceinline__ uint32_t lds_byte_off(const void* p) {
    // generic -> LDS(addrspace 3) cast yields the 32-bit LDS byte offset
    return (uint32_t)(uintptr_t)(const __attribute__((address_space(3))) char*)p;
}

__device__ __forceinline__ void async_ld_b128(uint32_t lds_byte, const float* g) {
    // copies 16 bytes from global address g to LDS byte offset lds_byte, no VGPR data, tracked by ASYNCcnt
    asm volatile("global_load_async_to_lds_b128 %0, %1, off"
                 :: "v"(lds_byte), "v"((uint64_t)(uintptr_t)g)
                 : "memory");
}

#define WAIT_ASYNC_LE16() asm volatile("s_wait_asynccnt 0x10" ::: "memory")
#define WAIT_ASYNC_0()    asm volatile("s_wait_asynccnt 0x0"  ::: "memory")

// ---------------------------------------------------------------------------
// Kernel 1: e_norms[j] = ||embedding[j]||^2   (one wave32 per codebook row)
// ---------------------------------------------------------------------------
__global__ void vq_enorm(const float* __restrict__ emb, float* __restrict__ enorm) {
    int gtid = blockIdx.x * blockDim.x + threadIdx.x;
    int row  = gtid >> 5;
    int lane = threadIdx.x & 31;
    if (row >= N_BINS) return;
    const float* r = emb + (size_t)row * D_MODEL;
    float s = 0.f;
    #pragma unroll 4
    for (int k = lane; k < D_MODEL; k += 32) { float v = r[k]; s += v * v; }
    #pragma unroll
    for (int off = 16; off > 0; off >>= 1) s += __shfl_xor(s, off, 32);
    if (lane == 0) enorm[row] = s;
}

// ---------------------------------------------------------------------------
// Kernel 2: fused  S = z @ E^T  (fp32 WMMA)  +  row-wise argmin of
//           d = ||e_j||^2 - 2*S_ij   (||z||^2 is constant per row).
//           B tiles double-buffered via async global->LDS copies.
// ---------------------------------------------------------------------------
__global__ void __launch_bounds__(WG_THREADS)
vq_argmin(const float* __restrict__ z, const float* __restrict__ emb,
          const float* __restrict__ enorm, int* __restrict__ idx_out) {
    extern __shared__ float lds[];
    float* At  = lds;                                  // [M_TILE][LDS_STRIDE]
    float* Bt0 = lds + M_TILE * LDS_STRIDE;            // [16][LDS_STRIDE] ping
    float* Bt1 = Bt0 + 16 * LDS_STRIDE;                // [16][LDS_STRIDE] pong

    const int t     = threadIdx.x;
    const int lane  = t & 31;
    const int wave  = t >> 5;                          // 0..3
    const int mrow0 = blockIdx.x * M_TILE;

    // ---- stage A tile: 64 rows x 512 f32 into LDS (float4, coalesced) ----
    {
        const float* src = z + (size_t)mrow0 * D_MODEL;
        #pragma unroll 4
        for (int i = 0; i < (M_TILE * D_MODEL / 4) / WG_THREADS; ++i) {
            int f = t + i * WG_THREADS;                // flat float4 id
            int d = f << 2;
            int r = d >> 9;                            // / 512
            int c = d & 511;
            *(v4f*)(At + r * LDS_STRIDE + c) = *(const v4f*)(src + r * D_MODEL + c);
        }
    }

    // async-issue one 16-row B tile (2048 float4 -> 16 b128 per thread)
    auto issue_B = [&](int nb, float* dstBase) {
        const float* src = emb + (size_t)nb * D_MODEL;
        uint32_t dst0 = lds_byte_off(dstBase);
        #pragma unroll
        for (int i = 0; i < (16 * D_MODEL / 4) / WG_THREADS; ++i) {
            int f = t + i * WG_THREADS;
            int d = f << 2;
            int r = d >> 9;
            int c = d & 511;
            async_ld_b128(dst0 + (uint32_t)(r * LDS_STRIDE + c) * 4u,
                          src + r * D_MODEL + c);
        }
    };

    issue_B(0, Bt0);                                   // prime the pipeline

    const int mwave = wave << 4;                       // this wave's 16-row sub-tile
    const int col   = lane & 15;                       // M index for A, N index for B
    const int khalf = (lane >> 4) << 1;                // K sub-offset 0 or 2

    float minval[8];
    int   minidx[8];
    #pragma unroll
    for (int v = 0; v < 8; ++v) { minval[v] = 3.4e38f; minidx[v] = 0; }

    const float* Arow = At + (mwave + col) * LDS_STRIDE + khalf;

    for (int nt = 0; nt < N_TILES; ++nt) {
        const int nb = nt << 4;
        float* Bcur  = (nt & 1) ? Bt1 : Bt0;
        float* Bnext = (nt & 1) ? Bt0 : Bt1;

        if (nt + 1 < N_TILES) {
            issue_B(nb + 16, Bnext);                   // overlap with this tile's math
            WAIT_ASYNC_LE16();                         // our tile-nt copies landed
        } else {
            WAIT_ASYNC_0();
        }
        __syncthreads();                               // all waves' copies visible
                                                       // (iter 0: also A-tile stores)

        float en = enorm[nb + col];                    // hoisted: hides under WMMAs
        const float* Brow = Bcur + col * LDS_STRIDE + khalf;

        // ---- 16x16 score tile via 128 x V_WMMA_F32_16X16X4_F32 ----
        v8f acc = (v8f){0.f, 0.f, 0.f, 0.f, 0.f, 0.f, 0.f, 0.f};
        #pragma unroll 8
        for (int kk = 0; kk < D_MODEL; kk += 4) {
            v2f a = *(const v2f*)(Arow + kk);          // A[m][kk+khalf .. +1]
            v2f b = *(const v2f*)(Brow + kk);          // E[n][kk+khalf .. +1]
            acc = __builtin_amdgcn_wmma_f32_16x16x4_f32(
                false, a, false, b, (short)0, acc, false, false);
        }

        // ---- fold into running argmin:  d = ||e_j||^2 - 2*S ----
        int j = nb + col;
        #pragma unroll
        for (int v = 0; v < 8; ++v) {
            float dv = en - 2.0f * acc[v];
            if (dv < minval[v]) { minval[v] = dv; minidx[v] = j; }
        }
        __syncthreads();                               // done reading Bcur before
    }                                                  // it is refilled at nt+2

    // ---- cross-lane argmin within each 16-lane half (wave32 butterflies) ----
    #pragma unroll
    for (int v = 0; v < 8; ++v) {
        float mv = minval[v];
        int   mi = minidx[v];
        #pragma unroll
        for (int off = 1; off < 16; off <<= 1) {
            float ov = __shfl_xor(mv, off, 32);
            int   oi = __shfl_xor(mi, off, 32);
            if (ov < mv || (ov == mv && oi < mi)) { mv = ov; mi = oi; }
        }
        if ((lane & 15) == 0) {
            // lanes 0..15 hold rows mwave+v ; lanes 16..31 hold rows mwave+8+v
            int row = mrow0 + mwave + v + ((lane >> 4) << 3);
            idx_out[row] = mi;
        }
    }
}

// ---------------------------------------------------------------------------
// Kernel 3: z_q = embedding[idx] (also z_q_st value) + per-row loss partial
// ---------------------------------------------------------------------------
__global__ void __launch_bounds__(128)
vq_gather_loss(const float* __restrict__ z, const float* __restrict__ emb,
               const int* __restrict__ idx, float* __restrict__ zq_out,
               float* __restrict__ partial) {
    int n = blockIdx.x;
    int t = threadIdx.x;
    int j = idx[n];
    const float* e  = emb + (size_t)j * D_MODEL;
    const float* zr = z   + (size_t)n * D_MODEL;
    float*       o  = zq_out + (size_t)n * D_MODEL;
    float s = 0.f;
    #pragma unroll
    for (int k = t; k < D_MODEL; k += 128) {
        float q = e[k];
        o[k] = q;
        float d = q - zr[k];
        s += d * d;
    }
    #pragma unroll
    for (int off = 16; off > 0; off >>= 1) s += __shfl_xor(s, off, 32);
    __shared__ float ws[4];
    if ((t & 31) == 0) ws[t >> 5] = s;
    __syncthreads();
    if (t == 0) partial[n] = ws[0] + ws[1] + ws[2] + ws[3];
}

// ---------------------------------------------------------------------------
// Kernel 4: deterministic reduction of 32768 partials -> loss scalar
//           loss = (1 + BETA) * mean((z_q - z)^2)
// ---------------------------------------------------------------------------
__global__ void __launch_bounds__(256)
vq_loss_final(const float* __restrict__ partial, float* __restrict__ loss_out) {
    __shared__ float red[256];
    int t = threadIdx.x;
    float s = 0.f;
    for (int i = t; i < N_VEC; i += 256) s += partial[i];   // fixed order
    red[t] = s;
    __syncthreads();
    #pragma unroll
    for (int off = 128; off > 0; off >>= 1) {
        if (t < off) red[t] += red[t + off];
        __syncthreads();
    }
    if (t == 0)
        *loss_out = (1.0f + BETA) * red[0] / (float)((size_t)N_VEC * D_MODEL);
}

// ---------------------------------------------------------------------------
extern "C" void kernel_launch(void* const* d_in, const int* in_sizes, int n_in,
                              void* d_out, int out_size, void* d_ws, size_t ws_size,
                              hipStream_t stream) {
    (void)in_sizes; (void)n_in; (void)out_size; (void)ws_size;

    const float* z   = (const float*)d_in[0];   // [8,4096,512] f32
    const float* emb = (const float*)d_in[1];   // [8192,512]   f32

    // d_out layout (flat, return order): z_q_st [16777216] f32, loss [1] f32,
    // indices [32768] int32
    float* zq   = (float*)d_out;
    float* loss = zq + (size_t)N_VEC * D_MODEL;
    int*   idx  = (int*)(loss + 1);

    // workspace: e_norms [8192] f32, loss partials [32768] f32
    float* enorm   = (float*)d_ws;
    float* partial = enorm + N_BINS;

    vq_enorm<<<(N_BINS * 32) / 256, 256, 0, stream>>>(emb, enorm);

    // LDS: A tile (64x516) + double-buffered B tiles (2 x 16x516) ~ 194 KB
    size_t ldsz = (size_t)(M_TILE + 32) * LDS_STRIDE * sizeof(float);
    vq_argmin<<<N_VEC / M_TILE, WG_THREADS, ldsz, stream>>>(z, emb, enorm, idx);

    vq_gather_loss<<<N_VEC, 128, 0, stream>>>(z, emb, idx, zq, partial);

    vq_loss_final<<<1, 256, 0, stream>>>(partial, loss);
}